// MonoGaussianAdapter_21466246546086
// MI455X (gfx1250) — compile-verified
//
#include <hip/hip_runtime.h>
#include <math.h>
#include <stdint.h>

// MonoGaussianAdapter for MI455X (gfx1250).
// Bandwidth-bound elementwise kernel (~231 MB traffic, ~10us floor @ 23.3 TB/s).
// gfx1250 path: per-wave async global->LDS staging of the 76B-stride AoS
// raw_gaussians. Each wave stages its 32 gaussians (608 dwords, 128B-aligned
// both in global and LDS) with 4x async_b128 + 3x async_b32 issues, then reads
// its lane's 19 floats from LDS (stride 19 is coprime with 64 banks -> no
// conflicts). Wave-private staging => only s_wait_asynccnt, no barrier.
// No WMMA by design: per-gaussian 3x3 algebra has no matmul reuse; HBM-bound.

#define THREADS 256
#define WAVES   (THREADS / 32)
#define FPG     19              // floats per raw gaussian (3 scale, 4 quat, 12 sh)
#define WAVE_DW (32 * FPG)      // 608 dwords staged per wave

__global__ __launch_bounds__(THREADS)
void mono_gauss_adapter_kernel(const float* __restrict__ ext,    // B*16
                               const float* __restrict__ intr,   // B*9
                               const float* __restrict__ coords, // G*2
                               const float* __restrict__ depths, // G
                               const float* __restrict__ opac,   // G
                               const float* __restrict__ rawg,   // G*19
                               const int*   __restrict__ hptr,   // scalar
                               const int*   __restrict__ wptr,   // scalar
                               float* __restrict__ out,          // 32*G
                               int N, int G)
{
    __shared__ float lds[WAVES * WAVE_DW];   // 8 waves * 608 dwords = 19456 B

    const int tid  = threadIdx.x;
    const int lane = tid & 31;
    const int wv   = tid >> 5;
    const int gid  = blockIdx.x * THREADS + tid;
    const int g0   = blockIdx.x * THREADS + wv * 32;   // wave's first gaussian

    // ---- async-stage this wave's 32 gaussians into its private LDS region ----
    const unsigned ldsBase = (unsigned)(uintptr_t)&lds[wv * WAVE_DW]; // low 32b of AS(3) addr = LDS byte offset
    const unsigned srcBase = (unsigned)g0 * FPG;                      // dword index into rawg

    if (g0 + 32 <= G) {
        // Fast path (always taken for G % 256 == 0): wave base is 128B-aligned
        // in global (2432B/wave = 19*128) and in LDS (2432B regions).
        // 608 dwords = 4 * (32 lanes * 4dw) + 3 * (32 lanes * 1dw).
#pragma unroll
        for (int j = 0; j < 4; ++j) {
            const unsigned rel = (unsigned)(j * 128 + lane * 4);      // 16B-aligned
            asm volatile("global_load_async_to_lds_b128 %0, %1, %2"
                         :: "v"(ldsBase + rel * 4u),
                            "v"((srcBase + rel) * 4u),
                            "s"(rawg)
                         : "memory");
        }
#pragma unroll
        for (int j = 0; j < 3; ++j) {
            const unsigned rel = (unsigned)(512 + j * 32 + lane);
            asm volatile("global_load_async_to_lds_b32 %0, %1, %2"
                         :: "v"(ldsBase + rel * 4u),
                            "v"((srcBase + rel) * 4u),
                            "s"(rawg)
                         : "memory");
        }
    } else {
        // Tail path: per-dword clamped b32 staging (clamped data never consumed).
        const unsigned limit = (unsigned)G * FPG - 1u;
#pragma unroll
        for (int j = 0; j < FPG; ++j) {
            const unsigned rel = (unsigned)(j * 32 + lane);
            unsigned src = srcBase + rel;
            if (src > limit) src = limit;
            asm volatile("global_load_async_to_lds_b32 %0, %1, %2"
                         :: "v"(ldsBase + rel * 4u),
                            "v"(src * 4u),
                            "s"(rawg)
                         : "memory");
        }
    }
    asm volatile("s_wait_asynccnt 0" ::: "memory");   // wave-private region: no barrier needed

    if (gid >= G) return;

    // ---------------- per-batch (block-uniform -> scalar path) ----------------
    const int b = (blockIdx.x * THREADS) / N;   // blocks never straddle batches (N % 256 == 0)
    const float* E  = ext  + b * 16;
    const float* Kb = intr + b * 9;

    const float r00 = E[0], r01 = E[1],  r02 = E[2],  tx = E[3];
    const float r10 = E[4], r11 = E[5],  r12 = E[6],  ty = E[7];
    const float r20 = E[8], r21 = E[9],  r22 = E[10], tz = E[11];

    const float k00 = Kb[0], k01 = Kb[1], k02 = Kb[2];
    const float k10 = Kb[3], k11 = Kb[4], k12 = Kb[5];
    const float k20 = Kb[6], k21 = Kb[7], k22 = Kb[8];

    const float Wf = (float)(*wptr), Hf = (float)(*hptr);
    const float px = 1.0f / Wf, py = 1.0f / Hf;

    // multiplier = 0.1 * sum(inv2x2(K[:2,:2]) @ [px,py])
    const float det2 = k00 * k11 - k01 * k10;
    const float mult = 0.1f * ((k11 - k10) * px + (k00 - k01) * py) / det2;

    // Kinv (adjugate / det)
    const float A00 = k11*k22 - k12*k21;
    const float A01 = k02*k21 - k01*k22;
    const float A02 = k01*k12 - k02*k11;
    const float A10 = k12*k20 - k10*k22;
    const float A11 = k00*k22 - k02*k20;
    const float A12 = k02*k10 - k00*k12;
    const float A20 = k10*k21 - k11*k20;
    const float A21 = k01*k20 - k00*k21;
    const float A22 = k00*k11 - k01*k10;
    const float idet = 1.0f / (k00*A00 + k01*A10 + k02*A20);

    // ---------------- per-gaussian ----------------
    const float u     = coords[gid * 2 + 0];
    const float v     = coords[gid * 2 + 1];
    const float depth = depths[gid];
    const float op    = opac[gid];

    const float* g = &lds[wv * WAVE_DW + lane * FPG];  // stride 19 -> conflict-free banks

    // --- means: ray through pixel, normalized, rotated to world, scaled by depth ---
    float dx = (A00*u + A01*v + A02) * idet;
    float dy = (A10*u + A11*v + A12) * idet;
    float dz = (A20*u + A21*v + A22) * idet;
    const float rn = 1.0f / sqrtf(dx*dx + dy*dy + dz*dz);
    dx *= rn; dy *= rn; dz *= rn;
    const float wxd = r00*dx + r01*dy + r02*dz;
    const float wyd = r10*dx + r11*dy + r12*dz;
    const float wzd = r20*dx + r21*dy + r22*dz;
    const float mx = tx + wxd * depth;
    const float my = ty + wyd * depth;
    const float mz = tz + wzd * depth;

    // --- scales: 0.5 + 14.5*sigmoid(raw), then * depth * multiplier ---
    const float dm = depth * mult;
    const float sx = (0.5f + 14.5f / (1.0f + expf(-g[0]))) * dm;
    const float sy = (0.5f + 14.5f / (1.0f + expf(-g[1]))) * dm;
    const float sz = (0.5f + 14.5f / (1.0f + expf(-g[2]))) * dm;
    const float s20 = sx * sx, s21 = sy * sy, s22 = sz * sz;

    // --- normalized quaternion (w,x,y,z) ---
    float qw = g[3], qx = g[4], qy = g[5], qz = g[6];
    const float iq = 1.0f / (sqrtf(qw*qw + qx*qx + qy*qy + qz*qz) + 1e-8f);
    qw *= iq; qx *= iq; qy *= iq; qz *= iq;

    const float R00 = 1.0f - 2.0f*(qy*qy + qz*qz);
    const float R01 = 2.0f*(qx*qy - qw*qz);
    const float R02 = 2.0f*(qx*qz + qw*qy);
    const float R10 = 2.0f*(qx*qy + qw*qz);
    const float R11 = 1.0f - 2.0f*(qx*qx + qz*qz);
    const float R12 = 2.0f*(qy*qz - qw*qx);
    const float R20 = 2.0f*(qx*qz - qw*qy);
    const float R21 = 2.0f*(qy*qz + qw*qx);
    const float R22 = 1.0f - 2.0f*(qx*qx + qy*qy);

    // --- cov1 = R diag(s^2) R^T (symmetric) ---
    const float C00 = R00*R00*s20 + R01*R01*s21 + R02*R02*s22;
    const float C01 = R00*R10*s20 + R01*R11*s21 + R02*R12*s22;
    const float C02 = R00*R20*s20 + R01*R21*s21 + R02*R22*s22;
    const float C11 = R10*R10*s20 + R11*R11*s21 + R12*R12*s22;
    const float C12 = R10*R20*s20 + R11*R21*s21 + R12*R22*s22;
    const float C22 = R20*R20*s20 + R21*R21*s21 + R22*R22*s22;

    // --- cov = c2w * cov1 * c2w^T ---
    const float T00 = r00*C00 + r01*C01 + r02*C02;
    const float T01 = r00*C01 + r01*C11 + r02*C12;
    const float T02 = r00*C02 + r01*C12 + r02*C22;
    const float T10 = r10*C00 + r11*C01 + r12*C02;
    const float T11 = r10*C01 + r11*C11 + r12*C12;
    const float T12 = r10*C02 + r11*C12 + r12*C22;
    const float T20 = r20*C00 + r21*C01 + r22*C02;
    const float T21 = r20*C01 + r21*C11 + r22*C12;
    const float T22 = r20*C02 + r21*C12 + r22*C22;

    const float V00 = T00*r00 + T01*r01 + T02*r02;
    const float V01 = T00*r10 + T01*r11 + T02*r12;
    const float V02 = T00*r20 + T01*r21 + T02*r22;
    const float V10 = T10*r00 + T11*r01 + T12*r02;
    const float V11 = T10*r10 + T11*r11 + T12*r12;
    const float V12 = T10*r20 + T11*r21 + T12*r22;
    const float V20 = T20*r00 + T21*r01 + T22*r02;
    const float V21 = T20*r10 + T21*r11 + T22*r12;
    const float V22 = T20*r20 + T21*r21 + T22*r22;

    // ---------------- stores (concat order: means|cov|harm|opac|scales|rot) ----------------
    const long long G64 = (long long)G;
    float* o_means = out;
    float* o_cov   = out + 3  * G64;
    float* o_harm  = out + 12 * G64;
    float* o_opac  = out + 24 * G64;
    float* o_scal  = out + 25 * G64;
    float* o_rot   = out + 28 * G64;

    o_means[gid*3 + 0] = mx;
    o_means[gid*3 + 1] = my;
    o_means[gid*3 + 2] = mz;

    float* cv = o_cov + (long long)gid * 9;
    cv[0] = V00; cv[1] = V01; cv[2] = V02;
    cv[3] = V10; cv[4] = V11; cv[5] = V12;
    cv[6] = V20; cv[7] = V21; cv[8] = V22;

    // --- harmonics: sh0 passthrough, sh1 = D1 @ (0.025*sh[:,1:4]); D1 = c2w[p][:,p], p=[1,2,0]
    const float m1 = 0.025f;   // 0.1 * 0.25^1
    float* hv = o_harm + (long long)gid * 12;
#pragma unroll
    for (int c = 0; c < 3; ++c) {
        const float h0 = g[7 + 4*c];
        const float e0 = g[8 + 4*c]  * m1;
        const float e1 = g[9 + 4*c]  * m1;
        const float e2 = g[10 + 4*c] * m1;
        const float h1 = r11*e0 + r12*e1 + r10*e2;
        const float h2 = r21*e0 + r22*e1 + r20*e2;
        const float h3 = r01*e0 + r02*e1 + r00*e2;
        *(float4*)(hv + 4*c) = make_float4(h0, h1, h2, h3);  // 16B aligned
    }

    o_opac[gid] = op;

    o_scal[gid*3 + 0] = sx;
    o_scal[gid*3 + 1] = sy;
    o_scal[gid*3 + 2] = sz;

    *(float4*)(o_rot + (long long)gid * 4) = make_float4(qw, qx, qy, qz);  // 16B aligned
}

extern "C" void kernel_launch(void* const* d_in, const int* in_sizes, int n_in,
                              void* d_out, int out_size, void* d_ws, size_t ws_size,
                              hipStream_t stream) {
    const float* ext    = (const float*)d_in[0];
    const float* intr   = (const float*)d_in[1];
    const float* coords = (const float*)d_in[2];
    const float* depths = (const float*)d_in[3];
    const float* opac   = (const float*)d_in[4];
    const float* rawg   = (const float*)d_in[5];
    const int*   hptr   = (const int*)d_in[6];
    const int*   wptr   = (const int*)d_in[7];
    float* out = (float*)d_out;

    const int B = in_sizes[0] / 16;     // extrinsics (B,1,4,4)
    const int G = in_sizes[3];          // depths (B,N) -> B*N
    const int N = G / B;

    const int blocks = (G + THREADS - 1) / THREADS;
    mono_gauss_adapter_kernel<<<blocks, THREADS, 0, stream>>>(
        ext, intr, coords, depths, opac, rawg, hptr, wptr, out, N, G);
}